// TD_BERT_558345748505
// MI455X (gfx1250) — compile-verified
//
#include <hip/hip_runtime.h>
#include <hip/hip_bf16.h>

typedef __attribute__((ext_vector_type(2))) float v2f;
typedef __attribute__((ext_vector_type(8))) float v8f;

#define B_SZ 256
#define S_SZ 512
#define D_SZ 1024
#define O_SZ 3
#define NPAD 16   // B-matrix N padded to 16 (cols 3..15 are zero)

// ---------------------------------------------------------------------------
// Kernel 0: build transposed, zero-padded weights Wt[16][1024] in workspace.
//   Wt[n][k] = (n < 3) ? fc_w[k*3 + n] : 0
// 16*1024 = 16384 elements -> 64 blocks x 256 threads.
// ---------------------------------------------------------------------------
__global__ __launch_bounds__(256)
void td_bert_wpad_kernel(const float* __restrict__ fc_w,
                         float* __restrict__ wt) {
    const int idx = blockIdx.x * 256 + threadIdx.x;   // 0 .. 16*1024-1
    const int n = idx >> 10;          // 0..15
    const int k = idx & (D_SZ - 1);   // 0..1023
    wt[idx] = (n < O_SZ) ? fc_w[k * O_SZ + n] : 0.0f;
}

// ---------------------------------------------------------------------------
// Kernel 1: per-batch mask reduction + ragged span max-pool.
// One block per batch row, 256 threads; span is 1..8 rows of 1024 f32, each
// thread max-reduces 4 contiguous floats (float4, fully coalesced).
// ---------------------------------------------------------------------------
__global__ __launch_bounds__(256)
void td_bert_pool_kernel(const float* __restrict__ embed,
                         const int* __restrict__ t_ids,
                         const int* __restrict__ left_ids,
                         float* __restrict__ pooled) {
    const int b   = blockIdx.x;
    const int tid = threadIdx.x;

    __shared__ int red[256];
    __shared__ int s_start, s_span;

    // Pack both counts in one int: left count in [15:0], target count in [31:16].
    const int base = b * S_SZ;
    int lsum = (left_ids[base + tid] != 0) + (left_ids[base + tid + 256] != 0);
    int tsum = (t_ids[base + tid]    != 0) + (t_ids[base + tid + 256]    != 0);
    red[tid] = lsum + (tsum << 16);
    __syncthreads();

    #pragma unroll
    for (int off = 128; off > 0; off >>= 1) {
        if (tid < off) red[tid] += red[tid + off];
        __syncthreads();
    }

    if (tid == 0) {
        const int comb       = red[0];
        const int left_cnt   = comb & 0xFFFF;
        const int target_cnt = comb >> 16;
        s_start = left_cnt - 1;          // left_len
        s_span  = target_cnt - 2;        // target_len, in [1, 8]
    }
    __syncthreads();

    const int start = s_start;
    const int span  = s_span;

    // Each thread owns 4 contiguous features (D = 1024 = 256 * 4).
    const float4* erow =
        (const float4*)(embed + (size_t)(b * S_SZ + start) * D_SZ) + tid;
    float4 m = erow[0];
    for (int s = 1; s < span; ++s) {
        float4 v = erow[(size_t)s * (D_SZ / 4)];
        m.x = fmaxf(m.x, v.x);
        m.y = fmaxf(m.y, v.y);
        m.z = fmaxf(m.z, v.z);
        m.w = fmaxf(m.w, v.w);
    }
    ((float4*)(pooled + (size_t)b * D_SZ))[tid] = m;
}

// ---------------------------------------------------------------------------
// Kernel 2: FC + tanh via V_WMMA_F32_16X16X4_F32.
// One wave (32 threads) per 16-row tile of the batch; B=256 -> 16 waves.
// D = tanh(pooled[16x1024] x Wt^T + bias), K accumulated in steps of 4.
//
// VGPR layouts (ISA 7.12.2, 32-bit, wave32):
//   A 16x4 : lanes 0-15 -> M=lane, regs {K0,K1}; lanes 16-31 -> same M, {K2,K3}
//   B 4x16 : lanes 0-15 -> N=lane, regs {K0,K1}; lanes 16-31 -> N=lane-16, {K2,K3}
//   C 16x16: reg r, lane l: M = r + 8*(l>=16), N = l&15
// Both A and B fragments are one unconditional 8-byte load per lane per step.
// ---------------------------------------------------------------------------
__global__ __launch_bounds__(32)
void td_bert_fc_wmma_kernel(const float* __restrict__ pooled,
                            const float* __restrict__ wt,   // [16][1024] padded
                            const float* __restrict__ fc_b,
                            float* __restrict__ out) {
    const int tile = blockIdx.x;          // 0..15
    const int lane = threadIdx.x;         // 0..31
    const int half = lane >> 4;           // 0 or 1
    const int l    = lane & 15;

    const int row = tile * 16 + l;        // A-matrix row M for this lane
    const float* arow = pooled + (size_t)row * D_SZ + 2 * half;
    const float* brow = wt     + (size_t)l   * D_SZ + 2 * half;  // column l of W

    v8f c = {};

    #pragma unroll 8
    for (int k0 = 0; k0 < D_SZ; k0 += 4) {
        const float2 av = *(const float2*)(arow + k0);
        const float2 bv = *(const float2*)(brow + k0);
        v2f a;    a.x    = av.x; a.y    = av.y;
        v2f bmat; bmat.x = bv.x; bmat.y = bv.y;
        // 8 args: (neg_a, A, neg_b, B, c_mod, C, reuse_a, reuse_b)
        c = __builtin_amdgcn_wmma_f32_16x16x4_f32(
            false, a, false, bmat, (short)0, c, false, false);
    }

    // Store: only the first O_SZ=3 output columns are real.
    if (l < O_SZ) {
        const float bias = fc_b[l];
        #pragma unroll
        for (int r = 0; r < 8; ++r) {
            const int m = r + 8 * half;
            out[(size_t)(tile * 16 + m) * O_SZ + l] = tanhf(c[r] + bias);
        }
    }
}

extern "C" void kernel_launch(void* const* d_in, const int* in_sizes, int n_in,
                              void* d_out, int out_size, void* d_ws, size_t ws_size,
                              hipStream_t stream) {
    const float* embed    = (const float*)d_in[0];   // [B,S,D] f32
    const int*   t_ids    = (const int*)d_in[1];     // [B,S]  i32
    const int*   left_ids = (const int*)d_in[2];     // [B,S]  i32
    const float* fc_w     = (const float*)d_in[3];   // [D,O]  f32
    const float* fc_b     = (const float*)d_in[4];   // [O]    f32
    float*       out      = (float*)d_out;           // [B,O]  f32

    float* pooled = (float*)d_ws;                    // [B,D]   f32 (1 MB)
    float* wtpad  = pooled + (size_t)B_SZ * D_SZ;    // [16,D]  f32 (64 KB)

    td_bert_wpad_kernel<<<(NPAD * D_SZ) / 256, 256, 0, stream>>>(fc_w, wtpad);
    td_bert_pool_kernel<<<B_SZ, 256, 0, stream>>>(embed, t_ids, left_ids, pooled);
    td_bert_fc_wmma_kernel<<<B_SZ / 16, 32, 0, stream>>>(pooled, wtpad, fc_b, out);
}